// Graphormer_18992345383253
// MI455X (gfx1250) — compile-verified
//
#include <hip/hip_runtime.h>

// GCN (2-layer, PyG GCNConv semantics) for MI455X / gfx1250.
// Bandwidth-bound problem (~4 GB moved vs 10 GFLOP of GEMM), so GEMMs use the
// full-precision V_WMMA_F32_16X16X4_F32 tensor path (fp32 reference parity),
// and the edge aggregation is a wave-per-edge float4 gather + fp32 atomic scatter.

typedef __attribute__((ext_vector_type(2))) float v2f;
typedef __attribute__((ext_vector_type(8))) float v8f;

#define HDIM 128  // F_IN == H == 128 in this problem

// ---- degree / normalization -------------------------------------------------

__global__ void deg_init_kernel(float* __restrict__ deg, int n) {
  int i = blockIdx.x * blockDim.x + threadIdx.x;
  if (i < n) deg[i] = 1.0f;  // self-loop contribution
}

__global__ void deg_accum_kernel(const long long* __restrict__ dst,
                                 float* __restrict__ deg, int e) {
  int i = blockIdx.x * blockDim.x + threadIdx.x;
  if (i < e) atomicAdd(&deg[(int)dst[i]], 1.0f);  // exact: counts < 2^24
}

__global__ void deg_to_dinv_kernel(float* __restrict__ deg, int n) {
  int i = blockIdx.x * blockDim.x + threadIdx.x;
  if (i < n) {
    float d = deg[i];
    deg[i] = (d > 0.0f) ? rsqrtf(d) : 0.0f;
  }
}

// ---- fp32 WMMA GEMM: Out[N,128] = X[N,128] @ W[128,128] (+ bias) ------------
// One wave -> one 16x16 C tile via V_WMMA_F32_16X16X4_F32 (K stepped by 4).
// A 16x4 fp32 layout: lane l holds row M=l%16; lanes 0-15 carry K={kk,kk+1},
// lanes 16-31 carry K={kk+2,kk+3} (VGPR0=even K, VGPR1=odd K). B 4x16 mirrors
// this with N=l%16. C/D: VGPR j holds M=j (lanes 0-15) / M=j+8 (lanes 16-31).
__global__ void gemm128_kernel(const float* __restrict__ X,
                               const float* __restrict__ W,
                               const float* __restrict__ bias,
                               float* __restrict__ Out) {
  const int lane  = threadIdx.x & 31;
  const int wave  = threadIdx.x >> 5;       // 8 waves: column tiles 0..7
  const int row0  = blockIdx.x << 4;        // 16 rows per block
  const int col0  = wave << 4;
  const int m     = lane & 15;
  const int khalf = (lane >> 4) << 1;       // 0 for lanes 0-15, 2 for 16-31

  const float* __restrict__ xrow = X + (size_t)(row0 + m) * HDIM;
  const float* __restrict__ wcol = W + (col0 + m);

  v8f acc = {};
#pragma unroll
  for (int kk = 0; kk < HDIM; kk += 4) {
    const int k0 = kk + khalf;
    v2f a, b;
    a.x = xrow[k0];
    a.y = xrow[k0 + 1];
    b.x = wcol[(size_t)k0 * HDIM];
    b.y = wcol[(size_t)(k0 + 1) * HDIM];
    // (neg_a, A, neg_b, B, c_mod, C, reuse_a, reuse_b)
    acc = __builtin_amdgcn_wmma_f32_16x16x4_f32(false, a, false, b,
                                                (short)0, acc, false, false);
  }

  const int   col   = col0 + m;
  const float bv    = bias ? bias[col] : 0.0f;
  const int   rbase = row0 + ((lane >> 4) << 3);
#pragma unroll
  for (int j = 0; j < 8; ++j) {
    Out[(size_t)(rbase + j) * HDIM + col] = acc[j] + bv;
  }
}

// ---- self-loop init: agg[i,:] = g[i,:] * dinv[i]^2 --------------------------
// Thread per (node, float4 slice); fully initializes the scatter target.
__global__ void self_init_kernel(const float* __restrict__ G,
                                 const float* __restrict__ dinv,
                                 float* __restrict__ Agg, int n) {
  int t = blockIdx.x * blockDim.x + threadIdx.x;  // n*32 threads
  if (t >= n * 32) return;
  int node = t >> 5;
  int seg  = t & 31;
  float w = dinv[node];
  w *= w;
  float4 v = ((const float4*)(G + (size_t)node * HDIM))[seg];
  v.x *= w; v.y *= w; v.z *= w; v.w *= w;
  ((float4*)(Agg + (size_t)node * HDIM))[seg] = v;
}

// ---- edge scatter: agg[d,:] += g[s,:] * dinv[s]*dinv[d] ---------------------
// One wave per edge: 32 lanes x float4 = the full 128-float feature row.
__global__ void edge_scatter_kernel(const long long* __restrict__ src,
                                    const long long* __restrict__ dst,
                                    const float* __restrict__ dinv,
                                    const float* __restrict__ G,
                                    float* __restrict__ Agg, int e) {
  int edge = blockIdx.x * 8 + (threadIdx.x >> 5);  // 8 edges per 256-thr block
  if (edge >= e) return;
  int lane = threadIdx.x & 31;
  int s = (int)src[edge];
  int d = (int)dst[edge];
  float w = dinv[s] * dinv[d];
  float4 v = ((const float4*)(G + (size_t)s * HDIM))[lane];
  float* out = Agg + (size_t)d * HDIM + lane * 4;
  atomicAdd(out + 0, v.x * w);
  atomicAdd(out + 1, v.y * w);
  atomicAdd(out + 2, v.z * w);
  atomicAdd(out + 3, v.w * w);
}

// ---- epilogue: out = relu(agg + bias) ---------------------------------------

__global__ void bias_relu_kernel(const float* __restrict__ Agg,
                                 const float* __restrict__ bias,
                                 float* __restrict__ Out, int n) {
  int t = blockIdx.x * blockDim.x + threadIdx.x;  // n*32 threads
  if (t >= n * 32) return;
  int node = t >> 5;
  int seg  = t & 31;
  int c0   = seg << 2;
  float4 v = ((const float4*)(Agg + (size_t)node * HDIM))[seg];
  v.x = fmaxf(v.x + bias[c0 + 0], 0.0f);
  v.y = fmaxf(v.y + bias[c0 + 1], 0.0f);
  v.z = fmaxf(v.z + bias[c0 + 2], 0.0f);
  v.w = fmaxf(v.w + bias[c0 + 3], 0.0f);
  ((float4*)(Out + (size_t)node * HDIM))[seg] = v;
}

// ---- host orchestration -----------------------------------------------------

extern "C" void kernel_launch(void* const* d_in, const int* in_sizes, int n_in,
                              void* d_out, int out_size, void* d_ws, size_t ws_size,
                              hipStream_t stream) {
  (void)n_in; (void)out_size; (void)ws_size;

  const float*     x          = (const float*)d_in[0];
  const long long* edge_index = (const long long*)d_in[1];
  // d_in[2] = batch: unused (sorted equal-size graphs -> reshape is identity)
  const float*     W_in       = (const float*)d_in[3];
  const float*     b_in       = (const float*)d_in[4];
  const float*     Ws         = (const float*)d_in[5];
  const float*     bs         = (const float*)d_in[6];

  const int H    = in_sizes[4];          // 128
  const int F_IN = in_sizes[3] / H;      // 128
  const int N    = in_sizes[0] / F_IN;   // 102400
  const int E    = in_sizes[1] / 2;      // 1638400
  const int L    = in_sizes[6] / H;      // 2

  const long long* src = edge_index;      // edge_index[0, :]
  const long long* dst = edge_index + E;  // edge_index[1, :]

  // Workspace layout (fp32): dinv[N] | hbuf[N*H] | gbuf[N*H] | agg[N*H]
  float* ws   = (float*)d_ws;
  float* dinv = ws;
  float* hbuf = ws + N;
  float* gbuf = hbuf + (size_t)N * H;
  float* agg  = gbuf + (size_t)N * H;
  float* out  = (float*)d_out;

  const int thr = 256;

  // 1) symmetric normalization weights (depend only on edge_index)
  deg_init_kernel<<<(N + thr - 1) / thr, thr, 0, stream>>>(dinv, N);
  deg_accum_kernel<<<(E + thr - 1) / thr, thr, 0, stream>>>(dst, dinv, E);
  deg_to_dinv_kernel<<<(N + thr - 1) / thr, thr, 0, stream>>>(dinv, N);

  // 2) input transform: h0 = x @ W_in + b_in   (fp32 WMMA)
  gemm128_kernel<<<N / 16, thr, 0, stream>>>(x, W_in, b_in, hbuf);

  // 3) GCN layers: h = relu(scatter_norm(h @ W) + b)
  for (int l = 0; l < L; ++l) {
    gemm128_kernel<<<N / 16, thr, 0, stream>>>(hbuf, Ws + (size_t)l * H * H,
                                               nullptr, gbuf);
    self_init_kernel<<<(N * 32 + thr - 1) / thr, thr, 0, stream>>>(gbuf, dinv,
                                                                   agg, N);
    edge_scatter_kernel<<<(E + 7) / 8, thr, 0, stream>>>(src, dst, dinv, gbuf,
                                                         agg, E);
    bias_relu_kernel<<<(N * 32 + thr - 1) / thr, thr, 0, stream>>>(
        agg, bs + (size_t)l * H, (l == L - 1) ? out : hbuf, N);
  }
}